// LightGCN_42417097015620
// MI455X (gfx1250) — compile-verified
//
#include <hip/hip_runtime.h>
#include <stdint.h>
#include <stddef.h>

// ---------------------------------------------------------------------------
// LightGCN propagation for MI455X (gfx1250, wave32).
//
// CSR is built once on device, then 3 atomic-free gather-reduce SpMM layers.
// One wave32 per row; lane owns one float4 of the 128-dim embedding (512B
// coalesced row gathers). Edge (col,val) pairs are staged into LDS with
// double-buffered global_load_async_to_lds_b64 (ASYNCcnt, NT cache hint so
// the 51.2MB edge stream doesn't evict the L2-resident x matrices), values
// broadcast via v_readlane, gather rows warmed with global_prefetch_b8.
// Inner loop is a fixed fully-unrolled 32-edge chunk (tail vals zeroed) with
// even/odd split accumulators for 8 independent FMA chains.
// ---------------------------------------------------------------------------

#define DIM   128
#define DIM4  32   // float4 per row
#define WAVES_PER_BLOCK 8

// ---- CDNA5 helpers --------------------------------------------------------

__device__ __forceinline__ uint32_t lds_off(const void* p) {
  // Generic (flat) LDS addresses carry the aperture in bits [63:32];
  // bits [31:0] are the workgroup-relative LDS byte offset (ISA 10.2).
  return (uint32_t)(uintptr_t)p;
}

__device__ __forceinline__ void async_copy_b64_nt(uint32_t lds_byte_addr, const void* gaddr) {
  // Each lane copies 8 bytes global -> LDS. Tracked by ASYNCcnt.
  // NT: streamed edge data must not evict L2-resident embeddings.
  asm volatile("global_load_async_to_lds_b64 %0, %1, off th:TH_LOAD_NT"
               :: "v"(lds_byte_addr), "v"(gaddr)
               : "memory");
}

__device__ __forceinline__ void wait_async_le1() {
  asm volatile("s_wait_asynccnt 0x1" ::: "memory");
}
__device__ __forceinline__ void wait_async_0() {
  asm volatile("s_wait_asynccnt 0x0" ::: "memory");
}

// ---- CSR build ------------------------------------------------------------

__global__ void zero_i32(int* __restrict__ p, int n) {
  for (int i = blockIdx.x * blockDim.x + threadIdx.x; i < n; i += gridDim.x * blockDim.x)
    p[i] = 0;
}

__global__ void count_rows(const int* __restrict__ rows, int* __restrict__ cnt, int n) {
  for (int i = blockIdx.x * blockDim.x + threadIdx.x; i < n; i += gridDim.x * blockDim.x) {
    int r = __builtin_nontemporal_load(&rows[i]);
    atomicAdd(&cnt[r], 1);
  }
}

// Single-block exclusive scan (n ~ 100k, 98 chunks of 1024 -> trivial cost).
__global__ void __launch_bounds__(1024)
scan_excl(const int* __restrict__ cnt, int* __restrict__ ptr, int* __restrict__ cur, int n) {
  __shared__ int sm[1024];
  __shared__ int carry;
  if (threadIdx.x == 0) carry = 0;
  __syncthreads();
  for (int base = 0; base < n; base += 1024) {
    int i = base + (int)threadIdx.x;
    int v = (i < n) ? cnt[i] : 0;
    int val = v;
    sm[threadIdx.x] = val;
    __syncthreads();
    for (int d = 1; d < 1024; d <<= 1) {
      int t = (threadIdx.x >= (unsigned)d) ? sm[threadIdx.x - d] : 0;
      __syncthreads();
      val += t;
      sm[threadIdx.x] = val;
      __syncthreads();
    }
    int excl = val - v;
    int c = carry;
    if (i < n) { ptr[i] = c + excl; cur[i] = c + excl; }
    __syncthreads();
    if (threadIdx.x == 1023) carry = c + sm[1023];
    __syncthreads();
  }
  if (threadIdx.x == 0) ptr[n] = carry;
}

// Scatter COO edges into interleaved (col, val_bits) CSR slots.
// Packed as u64: low 32 = col, high 32 = val bits (little-endian int2 layout).
__global__ void build_cv(const int* __restrict__ rows, const int* __restrict__ cols,
                         const float* __restrict__ vals, int* __restrict__ cursor,
                         unsigned long long* __restrict__ cv, int n) {
  for (int i = blockIdx.x * blockDim.x + threadIdx.x; i < n; i += gridDim.x * blockDim.x) {
    int r = __builtin_nontemporal_load(&rows[i]);
    int c = __builtin_nontemporal_load(&cols[i]);
    float v = __builtin_nontemporal_load(&vals[i]);
    int p = atomicAdd(&cursor[r], 1);
    unsigned long long t =
        (unsigned long long)(unsigned)c |
        ((unsigned long long)(unsigned)__float_as_int(v) << 32);
    __builtin_nontemporal_store(t, &cv[p]);
  }
}

// ---- init / finalize ------------------------------------------------------

__global__ void init_copy(const float* __restrict__ emb, float* __restrict__ xa,
                          float* __restrict__ out, int n4) {
  const float4* e4 = (const float4*)emb;
  float4* a4 = (float4*)xa;
  float4* o4 = (float4*)out;
  for (int i = blockIdx.x * blockDim.x + threadIdx.x; i < n4; i += gridDim.x * blockDim.x) {
    float4 v = e4[i];
    a4[i] = v;
    o4[i] = v;
  }
}

__global__ void scale_out(float* __restrict__ out, int n4) {
  float4* o4 = (float4*)out;
  for (int i = blockIdx.x * blockDim.x + threadIdx.x; i < n4; i += gridDim.x * blockDim.x) {
    float4 v = o4[i];
    v.x *= 0.25f; v.y *= 0.25f; v.z *= 0.25f; v.w *= 0.25f;
    o4[i] = v;
  }
}

// ---- CSR SpMM: one wave per row, async LDS edge staging -------------------

__global__ void __launch_bounds__(256)
spmm_csr(const int* __restrict__ row_ptr,
         const unsigned long long* __restrict__ cv,
         const float* __restrict__ xin,
         float* __restrict__ xout,
         float* __restrict__ acc_out,
         int nrows, int nedges, int store_x) {
  __shared__ unsigned long long stage[WAVES_PER_BLOCK][2][32];

  const int lane = threadIdx.x & 31;
  const int wl   = threadIdx.x >> 5;
  const int row  = blockIdx.x * WAVES_PER_BLOCK + wl;
  if (row >= nrows) return;

  const int s = row_ptr[row];
  const int e = row_ptr[row + 1];
  const float4* xin4 = (const float4*)xin;

  float ax0 = 0.f, ay0 = 0.f, az0 = 0.f, aw0 = 0.f;
  float ax1 = 0.f, ay1 = 0.f, az1 = 0.f, aw1 = 0.f;

  if (s < e) {
    // prime chunk 0 (clamp so every lane reads inside cv[0..nedges-1];
    // any staged col is therefore a *valid* node id)
    {
      int gi = s + lane; if (gi > nedges - 1) gi = nedges - 1;
      async_copy_b64_nt(lds_off(&stage[wl][0][lane]), &cv[gi]);
    }
    int buf = 0;
    for (int base = s; base < e; base += 32) {
      const int next = base + 32;
      if (next < e) {
        int gi = next + lane; if (gi > nedges - 1) gi = nedges - 1;
        async_copy_b64_nt(lds_off(&stage[wl][buf ^ 1][lane]), &cv[gi]);
        wait_async_le1();       // current chunk landed; next still in flight
      } else {
        wait_async_0();
      }

      unsigned long long cvl = stage[wl][buf][lane];
      int c = (int)(unsigned)(cvl & 0xffffffffull);
      // Zero tail-lane values: cols are valid everywhere, so a fixed
      // fully-unrolled 32-edge loop is correct (0 * x contributes nothing).
      float v = (base + lane < e) ? __int_as_float((int)(cvl >> 32)) : 0.f;

      // Warm the 32 gather rows (L2-resident) before the broadcast loop.
      __builtin_prefetch(&xin4[(size_t)c * DIM4], 0, 1);

#pragma unroll
      for (int j = 0; j < 32; j += 2) {
        int    c0 = __builtin_amdgcn_readlane(c, j);
        float  v0 = __int_as_float(__builtin_amdgcn_readlane(__float_as_int(v), j));
        int    c1 = __builtin_amdgcn_readlane(c, j + 1);
        float  v1 = __int_as_float(__builtin_amdgcn_readlane(__float_as_int(v), j + 1));
        float4 x0 = xin4[(size_t)c0 * DIM4 + lane];   // 512B coalesced
        float4 x1 = xin4[(size_t)c1 * DIM4 + lane];
        ax0 = fmaf(v0, x0.x, ax0);
        ay0 = fmaf(v0, x0.y, ay0);
        az0 = fmaf(v0, x0.z, az0);
        aw0 = fmaf(v0, x0.w, aw0);
        ax1 = fmaf(v1, x1.x, ax1);
        ay1 = fmaf(v1, x1.y, ay1);
        az1 = fmaf(v1, x1.z, az1);
        aw1 = fmaf(v1, x1.w, aw1);
      }
      buf ^= 1;
    }
  }

  const float ax = ax0 + ax1, ay = ay0 + ay1, az = az0 + az1, aw = aw0 + aw1;
  const size_t o = (size_t)row * DIM4 + lane;
  if (store_x) {
    float4 r; r.x = ax; r.y = ay; r.z = az; r.w = aw;
    ((float4*)xout)[o] = r;
  }
  float4 a = ((float4*)acc_out)[o];
  a.x += ax; a.y += ay; a.z += az; a.w += aw;
  ((float4*)acc_out)[o] = a;
}

// ---- Fallback (small ws): atomic COO scatter ------------------------------

__global__ void zero_f32(float* __restrict__ p, int n4) {
  float4* p4 = (float4*)p;
  float4 z; z.x = z.y = z.z = z.w = 0.f;
  for (int i = blockIdx.x * blockDim.x + threadIdx.x; i < n4; i += gridDim.x * blockDim.x)
    p4[i] = z;
}

__global__ void __launch_bounds__(256)
edge_scatter(const int* __restrict__ rows, const int* __restrict__ cols,
             const float* __restrict__ vals, const float* __restrict__ xin,
             float* __restrict__ xout, int nE) {
  const int lane = threadIdx.x & 31;
  const int eid  = blockIdx.x * WAVES_PER_BLOCK + (threadIdx.x >> 5);
  if (eid >= nE) return;
  const int   r = rows[eid];
  const int   c = cols[eid];
  const float v = vals[eid];
  const float4 xv = ((const float4*)xin)[(size_t)c * DIM4 + lane];
  float* dst = &xout[(size_t)r * DIM + lane * 4];
  atomicAdd(dst + 0, v * xv.x);
  atomicAdd(dst + 1, v * xv.y);
  atomicAdd(dst + 2, v * xv.z);
  atomicAdd(dst + 3, v * xv.w);
}

__global__ void add_into(float* __restrict__ out, const float* __restrict__ x, int n4) {
  float4* o4 = (float4*)out;
  const float4* x4 = (const float4*)x;
  for (int i = blockIdx.x * blockDim.x + threadIdx.x; i < n4; i += gridDim.x * blockDim.x) {
    float4 a = o4[i]; float4 b = x4[i];
    a.x += b.x; a.y += b.y; a.z += b.z; a.w += b.w;
    o4[i] = a;
  }
}

// ---- Host-side launcher ---------------------------------------------------

static inline size_t align256(size_t x) { return (x + 255) & ~(size_t)255; }

extern "C" void kernel_launch(void* const* d_in, const int* in_sizes, int n_in,
                              void* d_out, int out_size, void* d_ws, size_t ws_size,
                              hipStream_t stream) {
  if (n_in < 4) return;
  const float* emb  = (const float*)d_in[0];
  const int*   rows = (const int*)  d_in[1];
  const int*   cols = (const int*)  d_in[2];
  const float* vals = (const float*)d_in[3];
  float* out = (float*)d_out;

  const int nNodes = in_sizes[0] / DIM;
  const int nE     = in_sizes[1];
  const int n4     = nNodes * DIM4;                 // float4 count of a matrix

  const size_t szPtr = align256((size_t)(nNodes + 1) * sizeof(int));
  const size_t szCV  = align256((size_t)nE * sizeof(unsigned long long));
  const size_t szX   = align256((size_t)nNodes * DIM * sizeof(float));
  const size_t needCSR = 3 * szPtr + szCV + 2 * szX;

  char* w = (char*)d_ws;

  const int T = 256;
  const int gN4   = (n4 + T - 1) / T;
  const int gE    = (nE + T - 1) / T;
  const int gRowW = (nNodes + WAVES_PER_BLOCK - 1) / WAVES_PER_BLOCK;  // wave/row
  const int gPtr  = (nNodes + 1 + T - 1) / T;

  if (ws_size >= needCSR) {
    int*  cnt    = (int*) (w);
    int*  rowptr = (int*) (w + szPtr);
    int*  cursor = (int*) (w + 2 * szPtr);
    unsigned long long* cv = (unsigned long long*)(w + 3 * szPtr);
    float* xa    = (float*)(w + 3 * szPtr + szCV);
    float* xb    = (float*)(w + 3 * szPtr + szCV + szX);

    zero_i32<<<gPtr, T, 0, stream>>>(cnt, nNodes + 1);
    count_rows<<<gE, T, 0, stream>>>(rows, cnt, nE);
    scan_excl<<<1, 1024, 0, stream>>>(cnt, rowptr, cursor, nNodes);
    build_cv<<<gE, T, 0, stream>>>(rows, cols, vals, cursor, cv, nE);
    init_copy<<<gN4, T, 0, stream>>>(emb, xa, out, n4);

    float* cur = xa;
    float* nxt = xb;
    for (int l = 0; l < 3; ++l) {
      const int store_x = (l < 2) ? 1 : 0;   // layer-3 x is never read again
      spmm_csr<<<gRowW, T, 0, stream>>>(rowptr, cv, cur, nxt, out, nNodes, nE, store_x);
      float* t = cur; cur = nxt; nxt = t;
    }
    scale_out<<<gN4, T, 0, stream>>>(out, n4);
  } else {
    // Atomic COO fallback (needs only 2 ping-pong buffers).
    float* xa = (float*)(w);
    float* xb = (float*)(w + szX);
    const int gEdgeW = (nE + WAVES_PER_BLOCK - 1) / WAVES_PER_BLOCK;

    init_copy<<<gN4, T, 0, stream>>>(emb, xa, out, n4);
    float* cur = xa;
    float* nxt = xb;
    for (int l = 0; l < 3; ++l) {
      zero_f32<<<gN4, T, 0, stream>>>(nxt, n4);
      edge_scatter<<<gEdgeW, T, 0, stream>>>(rows, cols, vals, cur, nxt, nE);
      add_into<<<gN4, T, 0, stream>>>(out, nxt, n4);
      float* t = cur; cur = nxt; nxt = t;
    }
    scale_out<<<gN4, T, 0, stream>>>(out, n4);
  }
}